// GCN_67774583930931
// MI455X (gfx1250) — compile-verified
//
#include <hip/hip_runtime.h>
#include <hip/hip_bf16.h>

// ---------------- problem constants (match reference) ----------------
#define IN_CH   128
#define HID_CH  128
#define OUT_CH  2
#define N_GRAPHS 64

typedef float v2f __attribute__((ext_vector_type(2)));
typedef float v8f __attribute__((ext_vector_type(8)));

// ---------------- workspace layout (float offsets) ----------------
#define OFF_DEG   0
#define OFF_H1    50048
#define OFF_AGG1  (OFF_H1   + 6400000)
#define OFF_H2    (OFF_AGG1 + 6400000)
#define OFF_AGG2  (OFF_H2   + 100000)
#define OFF_SUMS  (OFF_AGG2 + 100000)
#define OFF_CNT   (OFF_SUMS + 128)

// ---------------- degree / init kernels ----------------
__global__ void gcn_init_kernel(float* __restrict__ deg, float* __restrict__ sums,
                                float* __restrict__ cnt, int n) {
    int i = blockIdx.x * blockDim.x + threadIdx.x;
    if (i < n) deg[i] = 1.0f;                 // self-loop contributes 1 to in-degree
    if (i < N_GRAPHS * OUT_CH) sums[i] = 0.0f;
    if (i < N_GRAPHS) cnt[i] = 0.0f;
}

__global__ void gcn_degcount_kernel(const int* __restrict__ dst,
                                    float* __restrict__ deg, int e) {
    int i = blockIdx.x * blockDim.x + threadIdx.x;
    if (i < e) atomicAdd(&deg[dst[i]], 1.0f);
}

__global__ void gcn_rsqrt_kernel(float* __restrict__ deg, int n) {
    int i = blockIdx.x * blockDim.x + threadIdx.x;
    if (i < n) deg[i] = rsqrtf(deg[i]);       // deg now holds d^{-1/2}
}

// ---------------- GEMM1: H = X @ W1, fp32 WMMA 16x16x4 ----------------
// Block = 256 threads = 8 waves; computes a 16x128 output tile.
// X tile staged memory->LDS with GLOBAL_LOAD_ASYNC_TO_LDS_B128 (ASYNCcnt path).
// LDS row stride 132 floats: 528B rows -> 16B-aligned b128 targets, banks
// (4*m + k) mod 64 distinct for m=0..15 -> conflict-free A reads.
__global__ __launch_bounds__(256)
void gcn_gemm1_wmma_kernel(const float* __restrict__ X, const float* __restrict__ W,
                           float* __restrict__ H, int n) {
    __shared__ float xt[16][132];
    const int row_base = blockIdx.x * 16;

    // async copy of the 16x128 f32 tile: 512 x 16B chunks, 2 per thread
    for (int c = threadIdx.x; c < 512; c += 256) {
        int r = c >> 5;                       // tile row 0..15
        int j = c & 31;                       // 16B chunk within row
        int rr = row_base + r;
        if (rr >= n) rr = n - 1;              // clamp (never stored for OOB rows)
        unsigned ldsoff = (unsigned)(size_t)&xt[r][j * 4];
        const float* gptr = X + (size_t)rr * IN_CH + j * 4;
        asm volatile("global_load_async_to_lds_b128 %0, %1, off"
                     :: "v"(ldsoff), "v"(gptr) : "memory");
    }
    asm volatile("s_wait_asynccnt 0x0" ::: "memory");  // this wave's copies done
    __syncthreads();                                   // tile visible to all waves

    const int lane = threadIdx.x & 31;
    const int wv   = threadIdx.x >> 5;
    const int cb   = wv * 16;                 // output-column base for this wave
    const int m    = lane & 15;               // A: M index / B,D: N index
    const int kh   = (lane >> 4) << 1;        // K sub-offset: 0 (lanes 0-15), 2 (16-31)

    v8f acc = {0.f, 0.f, 0.f, 0.f, 0.f, 0.f, 0.f, 0.f};
    #pragma unroll 4
    for (int k0 = 0; k0 < IN_CH; k0 += 4) {
        v2f A, B;
        // A(16x4) element (m,k): reg k&1, lane m + 16*(k>>1)
        A.x = xt[m][k0 + kh + 0];
        A.y = xt[m][k0 + kh + 1];
        // B(4x16) element (k,nn): reg k&1, lane nn + 16*(k>>1)   (row-striped)
        B.x = W[(k0 + kh + 0) * HID_CH + cb + m];
        B.y = W[(k0 + kh + 1) * HID_CH + cb + m];
        acc = __builtin_amdgcn_wmma_f32_16x16x4_f32(
            /*neg_a=*/false, A, /*neg_b=*/false, B,
            /*c_mod=*/(short)0, acc, /*reuse_a=*/false, /*reuse_b=*/false);
    }

    // D(16x16): reg r -> M = r + 8*(lane>>4), N = lane&15
    const int rb = row_base + ((lane >> 4) << 3);
    const int nn = cb + m;
    if (row_base + 16 <= n) {                 // uniform fast path: no exec dance
        #pragma unroll
        for (int r = 0; r < 8; ++r)
            H[(rb + r) * HID_CH + nn] = acc[r];
    } else {
        #pragma unroll
        for (int r = 0; r < 8; ++r) {
            int row = rb + r;
            if (row < n) H[row * HID_CH + nn] = acc[r];
        }
    }
}

// -------- self-loop + bias init (float4): agg = h*dinv^2 + b --------
__global__ void gcn_selfinit1_kernel(const float* __restrict__ h1,
                                     const float* __restrict__ dinv,
                                     const float* __restrict__ b1,
                                     float* __restrict__ agg1, int n) {
    int t = blockIdx.x * blockDim.x + threadIdx.x;     // over n*32 float4s
    if (t >= n * (HID_CH / 4)) return;
    int i  = t >> 5;                                   // node
    int c4 = t & 31;                                   // float4 column
    float di = dinv[i];
    float s  = di * di;
    float4 h  = ((const float4*)h1)[t];
    float4 bb = ((const float4*)b1)[c4];
    float4 o;
    o.x = h.x * s + bb.x;  o.y = h.y * s + bb.y;
    o.z = h.z * s + bb.z;  o.w = h.w * s + bb.w;
    ((float4*)agg1)[t] = o;
}

// ---------------- edge scatter, layer 1: one wave per edge ----------------
__global__ void gcn_scatter1_kernel(const int* __restrict__ src,
                                    const int* __restrict__ dst,
                                    const float* __restrict__ dinv,
                                    const float* __restrict__ h1,
                                    float* __restrict__ agg1, int e) {
    int t = blockIdx.x * blockDim.x + threadIdx.x;
    int edge = t >> 5;
    if (edge >= e) return;
    int lane = t & 31;
    int s = src[edge], d = dst[edge];
    // prefetch a future edge's source row into cache (global_prefetch_b8)
    int ef = edge + 64;
    if (ef < e) {
        const float* pf = h1 + (size_t)src[ef] * HID_CH + lane * 4;
        __builtin_prefetch(pf, 0, 1);
    }
    float nrm = dinv[s] * dinv[d];
    float4 v = *((const float4*)(h1 + (size_t)s * HID_CH) + lane); // coalesced 512B/edge
    float* ap = agg1 + (size_t)d * HID_CH + lane * 4;
    atomicAdd(ap + 0, v.x * nrm);
    atomicAdd(ap + 1, v.y * nrm);
    atomicAdd(ap + 2, v.z * nrm);
    atomicAdd(ap + 3, v.w * nrm);
}

// ------- fused ReLU + GEMM2 (128->2) + layer-2 self-loop init: wave/node -------
__global__ __launch_bounds__(256)
void gcn_gemm2_kernel(const float* __restrict__ agg1, const float* __restrict__ W2,
                      const float* __restrict__ b2, const float* __restrict__ dinv,
                      float* __restrict__ h2, float* __restrict__ agg2, int n) {
    int wave_global = (blockIdx.x * blockDim.x + threadIdx.x) >> 5;
    if (wave_global >= n) return;
    int lane = threadIdx.x & 31;
    const float* row = agg1 + (size_t)wave_global * HID_CH;
    float s0 = 0.f, s1 = 0.f;
    #pragma unroll
    for (int c = lane; c < HID_CH; c += 32) {
        float v = fmaxf(row[c], 0.0f);        // ReLU folded into the load
        s0 += v * W2[c * OUT_CH + 0];
        s1 += v * W2[c * OUT_CH + 1];
    }
    #pragma unroll
    for (int off = 16; off > 0; off >>= 1) {
        s0 += __shfl_down(s0, off, 32);
        s1 += __shfl_down(s1, off, 32);
    }
    if (lane == 0) {
        h2[wave_global * OUT_CH + 0] = s0;
        h2[wave_global * OUT_CH + 1] = s1;
        float di = dinv[wave_global];
        float sl = di * di;
        agg2[wave_global * OUT_CH + 0] = s0 * sl + b2[0];
        agg2[wave_global * OUT_CH + 1] = s1 * sl + b2[1];
    }
}

// ---------------- edge scatter, layer 2: one thread per edge ----------------
__global__ void gcn_scatter2_kernel(const int* __restrict__ src,
                                    const int* __restrict__ dst,
                                    const float* __restrict__ dinv,
                                    const float* __restrict__ h2,
                                    float* __restrict__ agg2, int e) {
    int i = blockIdx.x * blockDim.x + threadIdx.x;
    if (i >= e) return;
    int s = src[i], d = dst[i];
    float nrm = dinv[s] * dinv[d];
    float2 v = *(const float2*)(h2 + (size_t)s * OUT_CH);
    atomicAdd(&agg2[(size_t)d * OUT_CH + 0], v.x * nrm);
    atomicAdd(&agg2[(size_t)d * OUT_CH + 1], v.y * nrm);
}

// ---------------- global mean pool: sums + counts ----------------
__global__ void gcn_pool_kernel(const int* __restrict__ batch,
                                const float* __restrict__ agg2,
                                float* __restrict__ sums, float* __restrict__ cnt,
                                int n) {
    int i = blockIdx.x * blockDim.x + threadIdx.x;
    if (i >= n) return;
    int g = batch[i];
    atomicAdd(&cnt[g], 1.0f);
    atomicAdd(&sums[g * OUT_CH + 0], agg2[i * OUT_CH + 0]);
    atomicAdd(&sums[g * OUT_CH + 1], agg2[i * OUT_CH + 1]);
}

__global__ void gcn_finalize_kernel(const float* __restrict__ sums,
                                    const float* __restrict__ cnt,
                                    float* __restrict__ out) {
    int g = threadIdx.x;
    if (g >= N_GRAPHS) return;
    float c = fmaxf(cnt[g], 1.0f);
    out[g * OUT_CH + 0] = sums[g * OUT_CH + 0] / c;
    out[g * OUT_CH + 1] = sums[g * OUT_CH + 1] / c;
}

// ---------------- host-side launcher ----------------
extern "C" void kernel_launch(void* const* d_in, const int* in_sizes, int n_in,
                              void* d_out, int out_size, void* d_ws, size_t ws_size,
                              hipStream_t stream) {
    const float* x     = (const float*)d_in[0];
    const int*   ei    = (const int*)  d_in[1];   // (2, E) row-major
    const int*   batch = (const int*)  d_in[2];
    const float* W1    = (const float*)d_in[3];
    const float* b1    = (const float*)d_in[4];
    const float* W2    = (const float*)d_in[5];
    const float* b2    = (const float*)d_in[6];
    float* out = (float*)d_out;

    const int n = in_sizes[0] / IN_CH;   // 50000
    const int e = in_sizes[1] / 2;       // 800000
    const int* src = ei;
    const int* dst = ei + e;

    float* ws    = (float*)d_ws;
    float* dinv  = ws + OFF_DEG;   // deg, then rsqrt'd in place
    float* h1    = ws + OFF_H1;
    float* agg1  = ws + OFF_AGG1;
    float* h2    = ws + OFF_H2;
    float* agg2  = ws + OFF_AGG2;
    float* sums  = ws + OFF_SUMS;
    float* cnt   = ws + OFF_CNT;

    const int B = 256;

    // degrees (recomputed from inputs every call -> deterministic pipeline)
    gcn_init_kernel<<<(n + B - 1) / B, B, 0, stream>>>(dinv, sums, cnt, n);
    gcn_degcount_kernel<<<(e + B - 1) / B, B, 0, stream>>>(dst, dinv, e);
    gcn_rsqrt_kernel<<<(n + B - 1) / B, B, 0, stream>>>(dinv, n);

    // layer 1
    gcn_gemm1_wmma_kernel<<<(n + 15) / 16, B, 0, stream>>>(x, W1, h1, n);
    gcn_selfinit1_kernel<<<((size_t)n * (HID_CH / 4) + B - 1) / B, B, 0, stream>>>(h1, dinv, b1, agg1, n);
    gcn_scatter1_kernel<<<((size_t)e * 32 + B - 1) / B, B, 0, stream>>>(src, dst, dinv, h1, agg1, e);

    // layer 2 (ReLU fused into GEMM2 load)
    gcn_gemm2_kernel<<<((size_t)n * 32 + B - 1) / B, B, 0, stream>>>(agg1, W2, b2, dinv, h2, agg2, n);
    gcn_scatter2_kernel<<<(e + B - 1) / B, B, 0, stream>>>(src, dst, dinv, h2, agg2, e);

    // mean pool
    gcn_pool_kernel<<<(n + B - 1) / B, B, 0, stream>>>(batch, agg2, sums, cnt, n);
    gcn_finalize_kernel<<<1, 64, 0, stream>>>(sums, cnt, out);
}